// AttentionDotProduct_82300163326290
// MI455X (gfx1250) — compile-verified
//
#include <hip/hip_runtime.h>
#include <hip/hip_bf16.h>

// ---------------------------------------------------------------------------
// Attention dot-product (Luong) — B=32, S=4096, D=1024, f32.
// Memory-bound: encoder = 512 MB. Single-pass flash-style online softmax:
//   pass1: per (b, S-chunk) block — stage 16x1024 tile to LDS (async-to-LDS),
//          score tile via V_WMMA_F32_16X16X4_F32, online softmax rescale,
//          weighted accumulation from the LDS-hot tile. Raw scores go to the
//          attention output region; (m, l, acc[1024]) partials to d_ws.
//   pass2: per b — combine chunk partials, write weighted_sum, normalize
//          attention in place.
// ---------------------------------------------------------------------------

typedef __attribute__((ext_vector_type(2))) float v2f;
typedef __attribute__((ext_vector_type(8))) float v8f;
typedef __attribute__((ext_vector_type(4))) int   v4i;

#define BQ 32
#define SQ 4096
#define DQ 1024
#define S_CHUNK 128
#define NCHUNK (SQ / S_CHUNK)      // 32 chunks
#define TILE_R 16
#define NT (S_CHUNK / TILE_R)      // 8 sub-tiles per block
#define TPAD 1028                  // LDS tile row stride (floats): 4 mod 64 banks, 16B aligned
#define NTHREADS 256
#define NWAVES 8
#define PART_STRIDE 1028           // floats per partial record (acc[1024], m, l, pad) 16B aligned

#if __has_builtin(__builtin_amdgcn_global_load_async_to_lds_b128)
#define USE_ASYNC 1
#else
#define USE_ASYNC 0
#endif

typedef __attribute__((address_space(1))) v4i* gptr_v4i;
typedef __attribute__((address_space(3))) v4i* lptr_v4i;

__device__ __forceinline__ void async_wait_zero() {
#if __has_builtin(__builtin_amdgcn_s_wait_asynccnt)
    __builtin_amdgcn_s_wait_asynccnt(0);
#else
    asm volatile("s_wait_asynccnt 0" ::: "memory");
#endif
}

__global__ __launch_bounds__(NTHREADS)
void attn_pass1(const float* __restrict__ hidden,   // [B, D]
                const float* __restrict__ enc,      // [B, S, D]
                float* __restrict__ att_raw,        // [B, S] raw scores (d_out + B*D)
                float* __restrict__ part)           // [B*NCHUNK][PART_STRIDE]
{
    __shared__ float h[DQ];
    __shared__ float tile[TILE_R * TPAD];
    __shared__ float wpart[NWAVES * TILE_R];

    const int tid  = threadIdx.x;
    const int wave = tid >> 5;
    const int lane = tid & 31;
    const int half = lane >> 4;     // 0: lanes 0-15 (K 0,1), 1: lanes 16-31 (K 2,3)
    const int row  = lane & 15;     // A-matrix M index
    const int b    = blockIdx.y;
    const int chunk = blockIdx.x;
    const int s0   = chunk * S_CHUNK;

    // load decoder hidden state into LDS
    for (int i = tid; i < DQ; i += NTHREADS) h[i] = hidden[(size_t)b * DQ + i];
    __syncthreads();

    float m = -3.0e38f;
    float l = 0.0f;
    float4 acc = make_float4(0.f, 0.f, 0.f, 0.f);

    const float* base = enc + ((size_t)b * SQ + s0) * DQ;

    for (int t = 0; t < NT; ++t) {
        const float* tg = base + (size_t)t * TILE_R * DQ;

        // ---- stage 16 rows (64KB) into LDS, fully coalesced (one row per step)
        for (int r = 0; r < TILE_R; ++r) {
            const float* src = tg + r * DQ + tid * 4;
            float* dst = &tile[r * TPAD + tid * 4];
#if USE_ASYNC
            float* srcnc = const_cast<float*>(src);
            __builtin_amdgcn_global_load_async_to_lds_b128(
                (gptr_v4i)srcnc, (lptr_v4i)dst, 0, 0);
#else
            *(float4*)dst = *(const float4*)src;
#endif
        }
#if USE_ASYNC
        async_wait_zero();
#endif
        __syncthreads();

        // ---- scores via WMMA f32 16x16x4: wave w covers K in [w*128, w*128+128)
        // A[m,k] = tile[m][k]; B[k,n] = h[k] for all n -> every D column = scores
        v8f C = {0.f, 0.f, 0.f, 0.f, 0.f, 0.f, 0.f, 0.f};
        const int kbase = wave * 128;
        for (int j = 0; j < 32; ++j) {
            const int k0 = kbase + j * 4 + 2 * half;
            const float2 av = *(const float2*)&tile[row * TPAD + k0];
            const float2 bv = *(const float2*)&h[k0];
            v2f a;  a.x = av.x;  a.y = av.y;
            v2f bb; bb.x = bv.x; bb.y = bv.y;
            C = __builtin_amdgcn_wmma_f32_16x16x4_f32(
                    false, a, false, bb, (short)0, C, false, false);
        }
        // column N=0 lives in lanes 0 (M=0..7) and 16 (M=8..15)
        if (lane == 0) {
            #pragma unroll
            for (int r2 = 0; r2 < 8; ++r2) wpart[wave * TILE_R + r2] = C[r2];
        } else if (lane == 16) {
            #pragma unroll
            for (int r2 = 0; r2 < 8; ++r2) wpart[wave * TILE_R + 8 + r2] = C[r2];
        }
        __syncthreads();

        // ---- every thread reduces the 8 wave partials -> 16 tile scores (LDS broadcast)
        float sc[TILE_R];
        #pragma unroll
        for (int i = 0; i < TILE_R; ++i) {
            float v = 0.f;
            #pragma unroll
            for (int w = 0; w < NWAVES; ++w) v += wpart[w * TILE_R + i];
            sc[i] = v;
        }

        // ---- online softmax update (identical on all threads, deterministic)
        float mt = sc[0];
        #pragma unroll
        for (int i = 1; i < TILE_R; ++i) mt = fmaxf(mt, sc[i]);
        const float mn = fmaxf(m, mt);
        const float scale = __expf(m - mn);
        float p[TILE_R];
        float ls = 0.f;
        #pragma unroll
        for (int i = 0; i < TILE_R; ++i) { p[i] = __expf(sc[i] - mn); ls += p[i]; }
        l = l * scale + ls;
        acc.x *= scale; acc.y *= scale; acc.z *= scale; acc.w *= scale;

        // ---- weighted accumulation from the LDS-resident tile (thread owns d = tid*4..+3)
        #pragma unroll
        for (int i = 0; i < TILE_R; ++i) {
            const float4 e = *(const float4*)&tile[i * TPAD + tid * 4];
            acc.x = fmaf(p[i], e.x, acc.x);
            acc.y = fmaf(p[i], e.y, acc.y);
            acc.z = fmaf(p[i], e.z, acc.z);
            acc.w = fmaf(p[i], e.w, acc.w);
        }
        m = mn;

        // ---- emit raw scores (normalized later in pass2)
        if (tid < TILE_R) {
            float v = 0.f;
            #pragma unroll
            for (int w = 0; w < NWAVES; ++w) v += wpart[w * TILE_R + tid];
            att_raw[(size_t)b * SQ + s0 + t * TILE_R + tid] = v;
        }

        __syncthreads();   // protect tile/wpart before next sub-tile
    }

    // ---- write chunk partials
    float* P = part + (size_t)(b * NCHUNK + chunk) * PART_STRIDE;
    *(float4*)&P[tid * 4] = acc;
    if (tid == 0) { P[DQ] = m; P[DQ + 1] = l; }
}

__global__ __launch_bounds__(NTHREADS)
void attn_pass2(float* __restrict__ out_ws,     // [B, D] weighted_sum (d_out)
                float* __restrict__ att,        // [B, S] raw scores -> normalized in place
                const float* __restrict__ part) // [B*NCHUNK][PART_STRIDE]
{
    const int b = blockIdx.x;
    const int tid = threadIdx.x;
    const float* P0 = part + (size_t)b * NCHUNK * PART_STRIDE;

    // global max / denominator (identical on all threads; fixed-order reduction)
    float M = -3.0e38f;
    #pragma unroll
    for (int c = 0; c < NCHUNK; ++c) M = fmaxf(M, P0[c * PART_STRIDE + DQ]);
    float wgt[NCHUNK];
    float L = 0.f;
    #pragma unroll
    for (int c = 0; c < NCHUNK; ++c) {
        const float w = __expf(P0[c * PART_STRIDE + DQ] - M);
        wgt[c] = w;
        L += w * P0[c * PART_STRIDE + DQ + 1];
    }
    const float invL = 1.0f / L;

    // combine chunk accumulators: thread owns d = tid*4..+3
    float4 acc = make_float4(0.f, 0.f, 0.f, 0.f);
    #pragma unroll
    for (int c = 0; c < NCHUNK; ++c) {
        const float4 a = *(const float4*)&P0[c * PART_STRIDE + tid * 4];
        acc.x = fmaf(wgt[c], a.x, acc.x);
        acc.y = fmaf(wgt[c], a.y, acc.y);
        acc.z = fmaf(wgt[c], a.z, acc.z);
        acc.w = fmaf(wgt[c], a.w, acc.w);
    }
    float4 o;
    o.x = acc.x * invL; o.y = acc.y * invL; o.z = acc.z * invL; o.w = acc.w * invL;
    *(float4*)&out_ws[(size_t)b * DQ + tid * 4] = o;

    // normalize attention scores in place
    for (int i = 0; i < SQ / NTHREADS; ++i) {
        const size_t idx = (size_t)b * SQ + i * NTHREADS + tid;
        att[idx] = __expf(att[idx] - M) * invL;
    }
}

extern "C" void kernel_launch(void* const* d_in, const int* in_sizes, int n_in,
                              void* d_out, int out_size, void* d_ws, size_t ws_size,
                              hipStream_t stream) {
    (void)in_sizes; (void)n_in; (void)out_size; (void)ws_size;
    const float* hidden = (const float*)d_in[0];   // [B, D]
    const float* enc    = (const float*)d_in[1];   // [B, S, D]
    float* out = (float*)d_out;                    // weighted_sum [B*D] ++ attention [B*S]
    float* att = out + BQ * DQ;
    float* part = (float*)d_ws;                    // needs B*NCHUNK*PART_STRIDE*4 = ~4.02 MB

    attn_pass1<<<dim3(NCHUNK, BQ), NTHREADS, 0, stream>>>(hidden, enc, att, part);
    attn_pass2<<<dim3(BQ), NTHREADS, 0, stream>>>(out, att, part);
}